// MyModel_61933428409467
// MI455X (gfx1250) — compile-verified
//
#include <hip/hip_runtime.h>
#include <math.h>

// ---------------------------------------------------------------------------
// MyModel: out[b] = sum_l FFT3(fftshift(B[b]) * k1[:,:,:,l]) *
//                         FFT3(fftshift(B[b]) * k2[:,:,:,l])
// Volumes are 64x32x32. FFTs are done as DFT matrix multiplies on the CDNA5
// matrix pipes via v_wmma_f32_16x16x4_f32 (fp32 precision, 4 real matmuls per
// complex matmul). Workspace layout (floats): 32 volumes (b in 0..15, s in
// {k1,k2}) of [re 65536][im 65536]  -> 16 MB required in d_ws.
// ---------------------------------------------------------------------------

typedef float v2f __attribute__((ext_vector_type(2)));
typedef float v8f __attribute__((ext_vector_type(8)));

#define VOLE  65536      // elements per volume (64*32*32)
#define VOLF  131072     // floats per volume (re + im)
#define LOOPN 30
#define PI_F  3.14159265358979323846f

static __device__ __forceinline__ v8f wmma_f32(v2f a, v2f b, v8f c) {
  // D = A(16x4,f32) x B(4x16,f32) + C(16x16,f32); NEG/CM left at 0 (F32 has
  // C-negate only per ISA, we negate A fragments in VALU instead).
  return __builtin_amdgcn_wmma_f32_16x16x4_f32(false, a, false, b,
                                               (short)0, c, false, false);
}

// ---------------------------------------------------------------------------
// prep: ws[v] = fftshift(B[b]) * kernel_s[:,:,:,l], imag = 0
// fftshift on dims (64,32,32) == index XOR with (32,16,16).
// ---------------------------------------------------------------------------
__global__ void prep_kernel(const float* __restrict__ B,
                            const float* __restrict__ K1,
                            const float* __restrict__ K2,
                            float* __restrict__ ws, int l) {
  int idx = blockIdx.x * blockDim.x + threadIdx.x;   // 0 .. 32*65536-1
  int v = idx >> 16;
  int p = idx & 0xFFFF;
  int b = v >> 1, s = v & 1;
  int i = p >> 10, j = (p >> 5) & 31, k = p & 31;
  int is = i ^ 32, js = j ^ 16, ks = k ^ 16;
  float bs = B[b * VOLE + is * 1024 + js * 32 + ks];
  const float* kk = s ? K2 : K1;
  float kv = kk[(size_t)p * LOOPN + l];
  float* base = ws + (size_t)v * VOLF;
  base[p]         = bs * kv;
  base[VOLE + p]  = 0.0f;
}

// ---------------------------------------------------------------------------
// FFT along last axis (k, N=32):  Y = X * W32   (X: 32-row chunks x 32 cols)
// One wave per 32x32 complex tile; 128 WMMAs per wave.
// ---------------------------------------------------------------------------
__global__ void fft_axis2_kernel(float* __restrict__ ws) {
  __shared__ float twr[1024], twi[1024];          // W32 (kk,n) row-major
  __shared__ float xs[4][2][1024];                // per-wave X stage (re,im)
  const int tid = threadIdx.x, lane = tid & 31, w = tid >> 5;

  for (int t = tid; t < 1024; t += 128) {
    int kk = t >> 5, n = t & 31;
    float sv, cv;
    __sincosf(-2.0f * PI_F * (float)((kk * n) & 31) / 32.0f, &sv, &cv);
    twr[t] = cv; twi[t] = sv;
  }
  __syncthreads();

  const int task  = blockIdx.x * 4 + w;           // 0..2047
  const int v     = task >> 6;
  const int chunk = task & 63;
  float* re = ws + (size_t)v * VOLF;
  float* im = re + VOLE;
  const int row0 = chunk * 32;

  for (int t = lane; t < 1024; t += 32) {
    xs[w][0][t] = re[row0 * 32 + t];
    xs[w][1][t] = im[row0 * 32 + t];
  }
  __syncthreads();

  const int half = lane >> 4, lr = lane & 15;
  v8f accr[2][2] = {}, acci[2][2] = {};

  for (int kb = 0; kb < 8; ++kb) {
    const int kc = kb * 4 + 2 * half;             // K index base for this lane
    v2f ar[2], ai[2], ain[2], br[2], bi[2];
    #pragma unroll
    for (int mi = 0; mi < 2; ++mi) {
      int row = mi * 16 + lr;
      ar[mi].x = xs[w][0][row * 32 + kc];  ar[mi].y = xs[w][0][row * 32 + kc + 1];
      ai[mi].x = xs[w][1][row * 32 + kc];  ai[mi].y = xs[w][1][row * 32 + kc + 1];
      ain[mi] = -ai[mi];
    }
    #pragma unroll
    for (int ni = 0; ni < 2; ++ni) {
      int col = ni * 16 + lr;
      br[ni].x = twr[kc * 32 + col];  br[ni].y = twr[(kc + 1) * 32 + col];
      bi[ni].x = twi[kc * 32 + col];  bi[ni].y = twi[(kc + 1) * 32 + col];
    }
    #pragma unroll
    for (int mi = 0; mi < 2; ++mi)
      #pragma unroll
      for (int ni = 0; ni < 2; ++ni) {
        accr[mi][ni] = wmma_f32(ar[mi],  br[ni], accr[mi][ni]);  // +Xr*Wr
        accr[mi][ni] = wmma_f32(ain[mi], bi[ni], accr[mi][ni]);  // -Xi*Wi
        acci[mi][ni] = wmma_f32(ar[mi],  bi[ni], acci[mi][ni]);  // +Xr*Wi
        acci[mi][ni] = wmma_f32(ai[mi],  br[ni], acci[mi][ni]);  // +Xi*Wr
      }
  }

  #pragma unroll
  for (int mi = 0; mi < 2; ++mi)
    #pragma unroll
    for (int ni = 0; ni < 2; ++ni)
      #pragma unroll
      for (int vv = 0; vv < 8; ++vv) {
        int r = row0 + mi * 16 + vv + 8 * half;
        int c = ni * 16 + lr;
        re[r * 32 + c] = accr[mi][ni][vv];
        im[r * 32 + c] = acci[mi][ni][vv];
      }
}

// ---------------------------------------------------------------------------
// FFT along middle axis (j, N=32):  Y = W32 * P per i-plane (plane contiguous)
// ---------------------------------------------------------------------------
__global__ void fft_axis1_kernel(float* __restrict__ ws) {
  __shared__ float twr[1024], twi[1024];
  __shared__ float ps[4][2][1024];
  const int tid = threadIdx.x, lane = tid & 31, w = tid >> 5;

  for (int t = tid; t < 1024; t += 128) {
    int kk = t >> 5, n = t & 31;
    float sv, cv;
    __sincosf(-2.0f * PI_F * (float)((kk * n) & 31) / 32.0f, &sv, &cv);
    twr[t] = cv; twi[t] = sv;
  }
  __syncthreads();

  const int task = blockIdx.x * 4 + w;            // 0..2047
  const int v = task >> 6;
  const int ip = task & 63;                       // i-plane
  float* re = ws + (size_t)v * VOLF;
  float* im = re + VOLE;
  const int off = ip * 1024;                      // plane (j*32+k) contiguous

  for (int t = lane; t < 1024; t += 32) {
    ps[w][0][t] = re[off + t];
    ps[w][1][t] = im[off + t];
  }
  __syncthreads();

  const int half = lane >> 4, lr = lane & 15;
  v8f accr[2][2] = {}, acci[2][2] = {};

  for (int kb = 0; kb < 8; ++kb) {
    const int kc = kb * 4 + 2 * half;
    v2f awr[2], awi[2], awin[2], bpr[2], bpi[2];
    #pragma unroll
    for (int mi = 0; mi < 2; ++mi) {              // A = W32, rows = output j
      int row = mi * 16 + lr;
      awr[mi].x = twr[row * 32 + kc];  awr[mi].y = twr[row * 32 + kc + 1];
      awi[mi].x = twi[row * 32 + kc];  awi[mi].y = twi[row * 32 + kc + 1];
      awin[mi] = -awi[mi];
    }
    #pragma unroll
    for (int ni = 0; ni < 2; ++ni) {              // B = P, rows = K (jj)
      int col = ni * 16 + lr;
      bpr[ni].x = ps[w][0][kc * 32 + col];  bpr[ni].y = ps[w][0][(kc + 1) * 32 + col];
      bpi[ni].x = ps[w][1][kc * 32 + col];  bpi[ni].y = ps[w][1][(kc + 1) * 32 + col];
    }
    #pragma unroll
    for (int mi = 0; mi < 2; ++mi)
      #pragma unroll
      for (int ni = 0; ni < 2; ++ni) {
        accr[mi][ni] = wmma_f32(awr[mi],  bpr[ni], accr[mi][ni]);
        accr[mi][ni] = wmma_f32(awin[mi], bpi[ni], accr[mi][ni]);
        acci[mi][ni] = wmma_f32(awr[mi],  bpi[ni], acci[mi][ni]);
        acci[mi][ni] = wmma_f32(awi[mi],  bpr[ni], acci[mi][ni]);
      }
  }

  #pragma unroll
  for (int mi = 0; mi < 2; ++mi)
    #pragma unroll
    for (int ni = 0; ni < 2; ++ni)
      #pragma unroll
      for (int vv = 0; vv < 8; ++vv) {
        int r = mi * 16 + vv + 8 * half;          // output j
        int c = ni * 16 + lr;                     // k
        re[off + r * 32 + c] = accr[mi][ni][vv];
        im[off + r * 32 + c] = acci[mi][ni][vv];
      }
}

// ---------------------------------------------------------------------------
// FFT along first axis (i, N=64): volume viewed as 64 x 1024, Y = W64 * V.
// One wave per 16-column slab: M=64 (4 tiles), K=64 (16 steps), N=16.
// ---------------------------------------------------------------------------
__global__ void fft_axis0_kernel(float* __restrict__ ws) {
  __shared__ float twr[4096], twi[4096];          // W64 row-major
  __shared__ float qs[4][2][1024];                // 64 rows x 16 cols per wave
  const int tid = threadIdx.x, lane = tid & 31, w = tid >> 5;

  for (int t = tid; t < 4096; t += 128) {
    int ii = t >> 6, jj = t & 63;
    float sv, cv;
    __sincosf(-2.0f * PI_F * (float)((ii * jj) & 63) / 64.0f, &sv, &cv);
    twr[t] = cv; twi[t] = sv;
  }
  __syncthreads();

  const int task = blockIdx.x * 4 + w;            // 0..2047
  const int v = task >> 6;
  const int nc = task & 63;                       // 16-col slab of 1024
  float* re = ws + (size_t)v * VOLF;
  float* im = re + VOLE;
  const int col0 = nc * 16;

  for (int t = lane; t < 1024; t += 32) {
    int r = t >> 4, c = t & 15;
    qs[w][0][t] = re[r * 1024 + col0 + c];
    qs[w][1][t] = im[r * 1024 + col0 + c];
  }
  __syncthreads();

  const int half = lane >> 4, lr = lane & 15;
  v8f accr[4] = {}, acci[4] = {};

  for (int kb = 0; kb < 16; ++kb) {
    const int kc = kb * 4 + 2 * half;
    v2f br, bi;                                   // B = V slab, rows = K (ii)
    br.x = qs[w][0][kc * 16 + lr];  br.y = qs[w][0][(kc + 1) * 16 + lr];
    bi.x = qs[w][1][kc * 16 + lr];  bi.y = qs[w][1][(kc + 1) * 16 + lr];
    #pragma unroll
    for (int mi = 0; mi < 4; ++mi) {              // A = W64, rows = output i
      int row = mi * 16 + lr;
      v2f awr, awi, awin;
      awr.x = twr[row * 64 + kc];  awr.y = twr[row * 64 + kc + 1];
      awi.x = twi[row * 64 + kc];  awi.y = twi[row * 64 + kc + 1];
      awin = -awi;
      accr[mi] = wmma_f32(awr,  br, accr[mi]);
      accr[mi] = wmma_f32(awin, bi, accr[mi]);
      acci[mi] = wmma_f32(awr,  bi, acci[mi]);
      acci[mi] = wmma_f32(awi,  br, acci[mi]);
    }
  }

  #pragma unroll
  for (int mi = 0; mi < 4; ++mi)
    #pragma unroll
    for (int vv = 0; vv < 8; ++vv) {
      int r = mi * 16 + vv + 8 * half;            // output i
      re[r * 1024 + col0 + lr] = accr[mi][vv];
      im[r * 1024 + col0 + lr] = acci[mi][vv];
    }
}

// ---------------------------------------------------------------------------
// mac: out[b] (+)= F1[b] * F2[b]  (complex pointwise; interleaved re,im out)
// ---------------------------------------------------------------------------
__global__ void mac_kernel(const float* __restrict__ ws,
                           float* __restrict__ out, int init) {
  int idx = blockIdx.x * blockDim.x + threadIdx.x;  // 0 .. 16*65536-1
  int b = idx >> 16, p = idx & 0xFFFF;
  const float* v1 = ws + (size_t)(b * 2) * VOLF;
  const float* v2 = v1 + VOLF;
  float ar = v1[p], ai = v1[VOLE + p];
  float cr = v2[p], ci = v2[VOLE + p];
  float pr = ar * cr - ai * ci;
  float pi = ar * ci + ai * cr;
  float* o = out + (size_t)idx * 2;
  if (init) { o[0] = pr;  o[1] = pi;  }
  else      { o[0] += pr; o[1] += pi; }
}

// ---------------------------------------------------------------------------
extern "C" void kernel_launch(void* const* d_in, const int* in_sizes, int n_in,
                              void* d_out, int out_size, void* d_ws, size_t ws_size,
                              hipStream_t stream) {
  const float* B  = (const float*)d_in[0];   // (16,64,32,32)
  const float* K1 = (const float*)d_in[1];   // (64,32,32,30)
  const float* K2 = (const float*)d_in[2];   // (64,32,32,30)
  float* out = (float*)d_out;                // (16,64,32,32) complex interleaved
  float* ws  = (float*)d_ws;                 // needs 32*131072*4 = 16 MB

  (void)in_sizes; (void)n_in; (void)out_size; (void)ws_size;

  for (int l = 0; l < LOOPN; ++l) {
    prep_kernel     <<<8192, 256, 0, stream>>>(B, K1, K2, ws, l);
    fft_axis2_kernel<<< 512, 128, 0, stream>>>(ws);
    fft_axis1_kernel<<< 512, 128, 0, stream>>>(ws);
    fft_axis0_kernel<<< 512, 128, 0, stream>>>(ws);
    mac_kernel      <<<4096, 256, 0, stream>>>(ws, out, l == 0 ? 1 : 0);
  }
}